// FPNSSDDecoder_23278722744706
// MI455X (gfx1250) — compile-verified
//
#include <hip/hip_runtime.h>

// FPN-SSD decoder for MI455X (gfx1250).
//
// Memory-bound kernel (~106 MB traffic, ~75 MFLOP total): the optimization
// target is the data path, not matrix ops. Uses the CDNA5 async global->LDS
// DMA (GLOBAL_LOAD_ASYNC_TO_LDS_B128/B32, ASYNCcnt) to stage each wave's
// 32-row x 340-byte tile into LDS with perfectly coalesced 16B-per-lane
// transfers, then each lane consumes its own (LDS-bank-conflict-free) row
// with a branchless, fully unrolled online softmax.

namespace {
constexpr int kN           = 262144;  // priors
constexpr int kClasses     = 81;
constexpr int kRow         = 85;      // 4 loc + 81 logits (floats per row)
constexpr int kWave        = 32;      // wave32 on CDNA5
constexpr int kWavesPerBlk = 4;
constexpr int kBlk         = kWave * kWavesPerBlk;   // 128 threads
constexpr int kTileBytes   = kWave * kRow * 4;       // 10880 B per wave tile
}

__global__ __launch_bounds__(kBlk) void fpnssd_decode_kernel(
    const float* __restrict__ input,     // (N, 85)
    const float* __restrict__ priors,    // (N, 4)
    const float* __restrict__ variance,  // (2,)
    const float* __restrict__ thrP,      // (1,)
    float* __restrict__ out)             // [boxes 4N | conf N | pred N | det 6N]
{
    __shared__ float smem[kWavesPerBlk * kWave * kRow];  // 43,520 B, LDS offset 0

    const int lane    = threadIdx.x & (kWave - 1);
    const int wave    = threadIdx.x >> 5;
    const int rowBase = blockIdx.x * kBlk + wave * kWave; // wave's first row
    const int gRow    = blockIdx.x * kBlk + threadIdx.x;  // this lane's row

    // ---- Escape the LDS object so the async-copy asm is known to write it.
    // Without this, smem never escapes, alias analysis proves the asm cannot
    // touch it, loads from the never-stored array fold to undef, and the whole
    // softmax/decode pipeline is dead-code-eliminated.
    {
        const float* esc = smem;
        asm volatile("" :: "v"(esc) : "memory");
    }

    // ---- Stage 32 rows (10,880 B) global -> LDS via CDNA5 async DMA ----
    // Tile base = input + rowBase*85 floats; rowBase%32==0 -> 10880B multiple,
    // so the tile base is 16B aligned. 21 x 512B (b128/lane) + 1 x 128B (b32).
    // INST_OFFSET is added to BOTH the global and LDS address (ISA 08 §4.4),
    // so one VGPR address pair serves all 21 b128 issues.
    const float* gbase = input + (size_t)rowBase * kRow;
    const unsigned gv16 = (unsigned)lane * 16u;
    const unsigned lv16 = (unsigned)(wave * kTileBytes) + gv16;
    #define A128(OFF) asm volatile(                                       \
        "global_load_async_to_lds_b128 %0, %1, %2 offset:" #OFF           \
        :: "v"(lv16), "v"(gv16), "s"(gbase) : "memory")
    A128(0);     A128(512);   A128(1024);  A128(1536);  A128(2048);
    A128(2560);  A128(3072);  A128(3584);  A128(4096);  A128(4608);
    A128(5120);  A128(5632);  A128(6144);  A128(6656);  A128(7168);
    A128(7680);  A128(8192);  A128(8704);  A128(9216);  A128(9728);
    A128(10240);
    #undef A128
    {
        const unsigned gv4 = (unsigned)lane * 4u;
        const unsigned lv4 = (unsigned)(wave * kTileBytes) + gv4;
        asm volatile("global_load_async_to_lds_b32 %0, %1, %2 offset:10752"
                     :: "v"(lv4), "v"(gv4), "s"(gbase) : "memory");
    }
    asm volatile("s_wait_asynccnt 0" ::: "memory");
    // Tiles are wave-private: no workgroup barrier needed.

    // ---- Per-lane row processing from LDS ----
    // Row stride 85 floats -> bank = (85*lane + j) % 64 = (21*lane + j) % 64,
    // gcd(21,64)=1 -> conflict-free across the wave.
    const float* row = &smem[(size_t)threadIdx.x * kRow];
    const float l0 = row[0], l1 = row[1], l2 = row[2], l3 = row[3];

    // Branchless online softmax: running max m, rescaled sum s; argmax with
    // strict '>' (first occurrence, matching jnp.argmax). Both update paths
    // share one transcendental: e = exp(-|x - m|).
    //   x >  m : s = s*e + 1, m = x, pred = j
    //   x <= m : s = s + e              (d==0 -> e=1 -> s += 1, correct)
    // conf = exp(m - m)/sum = 1/s.
    float m = row[4];
    float s = 1.0f;
    int   pred = 0;
    #pragma unroll
    for (int j = 1; j < kClasses; ++j) {
        const float x  = row[4 + j];
        const bool  gt = x > m;
        const float e  = __expf(-fabsf(x - m));
        s    = fmaf(gt ? s : 1.0f, e, gt ? 1.0f : s);  // gt: s*e+1, else: s+e
        m    = fmaxf(m, x);
        pred = gt ? j : pred;
    }
    const float conf = 1.0f / s;

    // ---- Box decode ----
    const float4 p  = ((const float4*)priors)[gRow];  // 16B coalesced
    const float  v0 = variance[0];
    const float  v1 = variance[1];
    const float  thr = thrP[0];

    const float cx = p.x + l0 * v0 * p.z;
    const float cy = p.y + l1 * v0 * p.w;
    const float w  = p.z * __expf(l2 * v1);
    const float h  = p.w * __expf(l3 * v1);
    const float xmin = cx - 0.5f * w;
    const float ymin = cy - 0.5f * h;
    const float xmax = xmin + w;
    const float ymax = ymin + h;

    const bool valid = (pred > 0) && (conf > thr);
    const float b0 = valid ? xmin : 0.0f;
    const float b1 = valid ? ymin : 0.0f;
    const float b2 = valid ? xmax : 0.0f;
    const float b3 = valid ? ymax : 0.0f;
    const float cf = valid ? conf : 0.0f;
    const float pf = valid ? (float)(pred - 1) : -1.0f;

    // ---- Outputs (return-order concat) ----
    ((float4*)out)[gRow] = make_float4(b0, b1, b2, b3);   // boxes: 16B stores
    out[4 * kN + gRow] = cf;                              // class_conf
    out[5 * kN + gRow] = pf;                              // class_pred (exact in f32)
    float2* det = (float2*)(out + 6 * (size_t)kN + (size_t)gRow * 6);
    det[0] = make_float2(b0, b1);                         // detections: 8B stores
    det[1] = make_float2(b2, b3);
    det[2] = make_float2(cf, pf);
}

extern "C" void kernel_launch(void* const* d_in, const int* in_sizes, int n_in,
                              void* d_out, int out_size, void* d_ws, size_t ws_size,
                              hipStream_t stream) {
    const float* input    = (const float*)d_in[0];  // (1, N, 85) f32
    const float* priors   = (const float*)d_in[1];  // (N, 4)     f32
    const float* variance = (const float*)d_in[2];  // (2,)       f32
    const float* thr      = (const float*)d_in[3];  // (1,)       f32

    const dim3 grid(kN / kBlk);   // 2048 blocks
    const dim3 block(kBlk);       // 128 threads = 4 wave32
    fpnssd_decode_kernel<<<grid, block, 0, stream>>>(input, priors, variance,
                                                     thr, (float*)d_out);
}